// ModuleSoftsplat_34041910788554
// MI455X (gfx1250) — compile-verified
//
#include <hip/hip_runtime.h>
#include <stdint.h>

// Problem shape (fixed by reference setup_inputs()).
constexpr int N_  = 8;
constexpr int C_  = 32;
constexpr int H_  = 288;
constexpr int W_  = 512;
constexpr int HW  = H_ * W_;       // 147456
constexpr int CHW = C_ * HW;       // 4718592
#define EPS_F 1e-22f

typedef uint32_t u32;
typedef __attribute__((ext_vector_type(4))) u32 u32x4;
typedef __attribute__((ext_vector_type(8))) u32 u32x8;

// ---------------------------------------------------------------------------
// Pass 1: one thread per source pixel (n, y, x).
// Scatter-adds input*e into d_out (32 ch) and e into norm (1 ch) at the 4
// bilinear corners via global_atomic_add_f32. Corner weights/masks hoisted
// out of the channel loop; strided channel walk prefetched ahead.
// ---------------------------------------------------------------------------
__global__ __launch_bounds__(256)
void softsplat_scatter_kernel(const float* __restrict__ inp,
                              const float* __restrict__ flow,
                              const float* __restrict__ metric,
                              float* __restrict__ out,
                              float* __restrict__ norm) {
    const int p = blockIdx.x * blockDim.x + threadIdx.x;
    if (p >= N_ * HW) return;

    const int n   = p / HW;
    const int pix = p - n * HW;        // y*W + x
    const int y   = pix / W_;
    const int x   = pix - y * W_;

    const float fx = flow[(n * 2 + 0) * HW + pix];
    const float fy = flow[(n * 2 + 1) * HW + pix];
    const float e  = expf(metric[n * HW + pix]);

    const float Xp  = (float)x + fx;
    const float Yp  = (float)y + fy;
    const float x0f = floorf(Xp);
    const float y0f = floorf(Yp);
    const int   x0  = (int)x0f;
    const int   y0  = (int)y0f;
    const int   x1  = x0 + 1;
    const int   y1  = y0 + 1;

    const float tx = Xp - x0f;                    // in [0,1)
    const float ty = Yp - y0f;
    const float wA = (1.0f - tx) * (1.0f - ty);   // (y0, x0)
    const float wB = tx * (1.0f - ty);            // (y0, x1)
    const float wC = (1.0f - tx) * ty;            // (y1, x0)
    const float wD = tx * ty;                     // (y1, x1)

    const bool vx0 = (x0 >= 0) && (x0 < W_);
    const bool vx1 = (x1 >= 0) && (x1 < W_);
    const bool vy0 = (y0 >= 0) && (y0 < H_);
    const bool vy1 = (y1 >= 0) && (y1 < H_);
    const bool vA = vx0 && vy0;
    const bool vB = vx1 && vy0;
    const bool vC = vx0 && vy1;
    const bool vD = vx1 && vy1;

    const int iA = y0 * W_ + x0;   // only used when valid
    const int iB = iA + 1;
    const int iC = iA + W_;
    const int iD = iC + 1;

    // Norm channel: scatter e * w.
    {
        float* nb = norm + n * HW;
        if (vA) atomicAdd(nb + iA, e * wA);
        if (vB) atomicAdd(nb + iB, e * wB);
        if (vC) atomicAdd(nb + iC, e * wC);
        if (vD) atomicAdd(nb + iD, e * wD);
    }

    // Color channels: scatter input * e * w. Channel stride is HW, so each
    // iteration is fully coalesced across the wave (lanes differ in x).
    const float* __restrict__ inBase  = inp + (size_t)n * CHW + pix;
    float* __restrict__       outBase = out + (size_t)n * CHW;

    const float eA = e * wA, eB = e * wB, eC = e * wC, eD = e * wD;

    #pragma unroll 4
    for (int c = 0; c < C_; ++c) {
        // gfx1250 global_prefetch_b8: pull the strided channel walk ahead
        // while the atomic issue stream keeps the VMEM pipe busy.
        const int pc = (c + 4 < C_) ? (c + 4) : (C_ - 1);
        __builtin_prefetch(inBase + (size_t)pc * HW, 0, 3);

        const float v = inBase[(size_t)c * HW];
        float* ob = outBase + (size_t)c * HW;
        if (vA) atomicAdd(ob + iA, v * eA);
        if (vB) atomicAdd(ob + iB, v * eB);
        if (vC) atomicAdd(ob + iC, v * eC);
        if (vD) atomicAdd(ob + iD, v * eD);
    }
}

// ---------------------------------------------------------------------------
// Pass 2: out[n,c,pix] /= (norm[n,pix] + EPS).
// Each block owns a 1024-pixel chunk of one image n and loops over all 32
// channels, so norm is fetched ONCE per pixel (not 32x) and the divide is
// done once per pixel. The chunk is DMA'd into LDS by the Tensor Data Mover
// (tensor_load_to_lds, 1-D tile descriptor per CDNA5 ISA ch.8), drained with
// s_wait_tensorcnt, consumed via a single ds_load_b128 per thread, then
// reused from registers across the channel loop.
// ---------------------------------------------------------------------------
__global__ __launch_bounds__(256)
void softsplat_normalize_kernel(float* __restrict__ out,
                                const float* __restrict__ norm) {
    constexpr int CHUNK = 1024;               // pixels per block (4 KB tile)
    __shared__ float shNorm[CHUNK];

    const int blocksPerN = HW / CHUNK;        // 144
    const int n     = blockIdx.x / blocksPerN;
    const int chunk = blockIdx.x - n * blocksPerN;
    const int pix0  = chunk * CHUNK;

    if (threadIdx.x == 0) {
        const uint64_t gaddr = (uint64_t)(uintptr_t)(norm + (size_t)n * HW + pix0);
        const u32 ldsOff = (u32)(uintptr_t)(&shNorm[0]);   // low 32 bits = LDS offset

        // D# group 0 (128b): count=1 | lds_addr | global_addr[56:0] | type=2
        u32x4 g0;
        g0.x = 1u;                                          // count=1, is_restore=0, gather off
        g0.y = ldsOff;                                      // lds_addr (bytes)
        g0.z = (u32)gaddr;                                  // global_addr[31:0]
        g0.w = ((u32)(gaddr >> 32) & 0x01FFFFFFu)           // global_addr[56:32]
               | (2u << 30);                                // type = 2 ("image")

        // D# group 1 (256b): data_size=4B, 1-D tile of CHUNK elements.
        u32x8 g1;
        g1[0] = 2u << 16;                                   // data_size=4B; mask/flags/pad = 0
        g1[1] = ((u32)CHUNK & 0xFFFFu) << 16;               // tensor_dim0[15:0] @ bit48
        g1[2] = ((u32)CHUNK >> 16) | (1u << 16);            // tensor_dim0[31:16], tensor_dim1=1
        g1[3] = (u32)CHUNK << 16;                           // tile_dim0 = CHUNK
        g1[4] = 0u;                                         // tile_dim1=0 (unused), tile_dim2=0
        g1[5] = (u32)CHUNK;                                 // tensor_dim0_stride[31:0]
        g1[6] = 0u;                                         // stride0[47:32], stride1[15:0]
        g1[7] = 0u;                                         // stride1[47:16]

        asm volatile("tensor_load_to_lds %0, %1"
                     :: "s"(g0), "s"(g1) : "memory");
        asm volatile("s_wait_tensorcnt 0x0" ::: "memory");
    }
    __syncthreads();

    // Explicit ds_load_b128 of this thread's 4 norm pixels (16B aligned).
    // Inline asm keeps it on the LDS-indexed path (ds_load, DScnt) and is
    // immune to the compiler reasoning about the TDM-written shared array.
    const int li = threadIdx.x * 4;
    const u32 ldsAddr = (u32)(uintptr_t)(&shNorm[li]);      // byte offset in LDS
    float4 nm4;
    asm volatile("ds_load_b128 %0, %1\n\t"
                 "s_wait_dscnt 0x0"
                 : "=v"(nm4) : "v"(ldsAddr));

    const float r0 = 1.0f / (nm4.x + EPS_F);
    const float r1 = 1.0f / (nm4.y + EPS_F);
    const float r2 = 1.0f / (nm4.z + EPS_F);
    const float r3 = 1.0f / (nm4.w + EPS_F);

    float* base = out + (size_t)n * CHW + (size_t)pix0 + li;  // 16B aligned
    #pragma unroll 4
    for (int c = 0; c < C_; ++c) {
        float4* p = reinterpret_cast<float4*>(base + (size_t)c * HW);
        float4 o = *p;
        o.x *= r0;
        o.y *= r1;
        o.z *= r2;
        o.w *= r3;
        *p = o;
    }
}

extern "C" void kernel_launch(void* const* d_in, const int* in_sizes, int n_in,
                              void* d_out, int out_size, void* d_ws, size_t ws_size,
                              hipStream_t stream) {
    const float* tenInput  = (const float*)d_in[0];   // (8,32,288,512)
    const float* tenFlow   = (const float*)d_in[1];   // (8,2,288,512)
    const float* tenMetric = (const float*)d_in[2];   // (8,1,288,512)
    float* out  = (float*)d_out;                      // (8,32,288,512)
    float* norm = (float*)d_ws;                       // (8,288,512) accumulator, 4.7 MB

    // Zero accumulators every call (deterministic; graph-capture legal).
    hipMemsetAsync(out,  0, (size_t)N_ * CHW * sizeof(float), stream);
    hipMemsetAsync(norm, 0, (size_t)N_ * HW  * sizeof(float), stream);

    const int pixels = N_ * HW;                        // 1,179,648
    softsplat_scatter_kernel<<<(pixels + 255) / 256, 256, 0, stream>>>(
        tenInput, tenFlow, tenMetric, out, norm);

    const int nBlocks = N_ * (HW / 1024);              // 1152
    softsplat_normalize_kernel<<<nBlocks, 256, 0, stream>>>(out, norm);
}